// FaceIndexMap_59665685676480
// MI455X (gfx1250) — compile-verified
//
#include <hip/hip_runtime.h>
#include <hip/hip_bf16.h>

// ---------------------------------------------------------------------------
// Z-buffer triangle rasterizer for MI455X (gfx1250, wave32).
//
//   Per-face quantities, affine in pixel coords, PRE-SCALED by 1/area
//   (area = w0+w1+w2 = per-face constant, > 0 for any drawable face):
//     w_e'(x,y) = (a_e*x + b_e*y + c_e) / area        (3 edge functions)
//     q(x,y)    = 1/NEAR - (sum_e w_e/z_e) / area     (= 10 - 1/zp)
//   zp = 1/den' and q = 10 - den', so:
//     inside      <=> min3(w0',w1',w2') > 0
//     zp > NEAR   <=> q > 0            -> folds into the same min-test
//     closer      <=> q < bestq        (10 - x is monotone decreasing)
//     zp < FAR    <=> folds into init bestq = 10 - 1/FAR = 9.99
//   Validity+update: min(min3(w0',w1',w2'), q) > 0  &&  q < bestq.
//   Back-facing / degenerate faces (area<=0) get a rejecting column w0'=-1.
//
//   Evaluating {w0',w1',w2',q} for 16 pixels x 16 faces is four
//   V_WMMA_F32_16X16X4_F32: A(16x4)=[x y 1 0], B(4x16)=coeff columns.
//   min3/min are emitted via inline asm to avoid the compiler's
//   v_max_num_f32 canonicalization around IEEE fminf.
// ---------------------------------------------------------------------------

typedef __attribute__((ext_vector_type(2))) float v2f;
typedef __attribute__((ext_vector_type(8))) float v8f;

#define IMG_S     256
#define NFACES    4096
#define NTILES    (NFACES / 16)   // 256 face tiles of 16
#define NQ        4               // w0', w1', w2', q
#define TSTR      (NQ * 64)       // 256 floats per tile
#define INV_FAR   0.01f           // 1/FAR
#define INV_NEAR  10.0f           // 1/NEAR
#define Q_INIT    (INV_NEAR - INV_FAR)   // 9.99f

__device__ __forceinline__ float min3f(float a, float b, float c) {
  float d;
  asm("v_min3_num_f32 %0, %1, %2, %3" : "=v"(d) : "v"(a), "v"(b), "v"(c));
  return d;
}
__device__ __forceinline__ float min2f(float a, float b) {
  float d;
  asm("v_min_num_f32 %0, %1, %2" : "=v"(d) : "v"(a), "v"(b));
  return d;
}

// ---------------------------------------------------------------------------
// Kernel 1: per-face affine coefficients scaled by 1/area, scattered
// pre-swizzled into the WMMA B-operand lane layout (mirrors the documented
// 32-bit A 16x4 layout: VGPR0 = K0 | K2, VGPR1 = K1 | K3), stored as
// [tile][q][lane][2] -> one coalesced global_load_b64 per B operand.
// ---------------------------------------------------------------------------
__global__ __launch_bounds__(256) void coeff_kernel(const float* __restrict__ faces,
                                                    float* __restrict__ ws) {
  int f = blockIdx.x * blockDim.x + threadIdx.x;
  if (f >= NFACES) return;
  const float* p = faces + f * 9;
  float x0 = p[0], y0 = p[1], z0 = p[2];
  float x1 = p[3], y1 = p[4], z1 = p[5];
  float x2 = p[6], y2 = p[7], z2 = p[8];

  // w_e = a*XP + b*YP + c
  float a0 = y1 - y2, b0 = x2 - x1, c0 = x1 * y2 - x2 * y1;
  float a1 = y2 - y0, b1 = x0 - x2, c1 = x2 * y0 - x0 * y2;
  float a2 = y0 - y1, b2 = x1 - x0, c2 = x0 * y1 - x1 * y0;

  float iz0 = 1.0f / z0, iz1 = 1.0f / z1, iz2 = 1.0f / z2;
  float ad = a0 * iz0 + a1 * iz1 + a2 * iz2;
  float bd = b0 * iz0 + b1 * iz1 + b2 * iz2;
  float cd = c0 * iz0 + c1 * iz1 + c2 * iz2;

  float area = c0 + c1 + c2;              // == w0+w1+w2 (x/y coeffs cancel)

  float av[NQ], bv[NQ], cv[NQ];
  if (area > 0.0f) {
    float s = 1.0f / area;
    av[0] = a0 * s; bv[0] = b0 * s; cv[0] = c0 * s;
    av[1] = a1 * s; bv[1] = b1 * s; cv[1] = c1 * s;
    av[2] = a2 * s; bv[2] = b2 * s; cv[2] = c2 * s;
    // q = 1/NEAR - den'  (negated reciprocal-depth, shifted)
    av[3] = -ad * s; bv[3] = -bd * s; cv[3] = INV_NEAR - cd * s;
  } else {
    // Back-facing or degenerate: rejecting column (w0' = -1 everywhere).
#pragma unroll
    for (int q = 0; q < NQ; ++q) { av[q] = 0.0f; bv[q] = 0.0f; cv[q] = 0.0f; }
    cv[0] = -1.0f;
  }

  int tile = f >> 4;
  int n    = f & 15;
  float* t = ws + tile * TSTR;
#pragma unroll
  for (int q = 0; q < NQ; ++q) {
    t[q * 64 + 2 * n + 0]      = av[q];   // K=0 row (x coefficient)
    t[q * 64 + 2 * n + 1]      = bv[q];   // K=1 row (y coefficient)
    t[q * 64 + 32 + 2 * n + 0] = cv[q];   // K=2 row (constant)
    t[q * 64 + 32 + 2 * n + 1] = 0.0f;    // K=3 row (padding)
  }
}

// ---------------------------------------------------------------------------
// Kernel 2: one wave per 16-pixel strip (4096 waves). 256 tiles x 4 WMMAs.
// C/D layout: lane L holds face column N = L&15; acc VGPR r holds pixel
// M = r (lanes 0-15) or r+8 (lanes 16-31).
// ---------------------------------------------------------------------------
__global__ __launch_bounds__(256) void raster_kernel(const float* __restrict__ coef,
                                                     int* __restrict__ out) {
  const int lane = threadIdx.x & 31;
  const int wav  = blockIdx.x * (blockDim.x >> 5) + (threadIdx.x >> 5);
  const int row  = wav >> 4;           // image row
  const int col0 = (wav & 15) << 4;    // strip start column
  const int n    = lane & 15;
  const bool hi  = lane >= 16;

  // A operand [x y 1 0]: lanes 0-15 supply K0/K1 (x,y), lanes 16-31 K2/K3.
  const float yp = (2.0f * (float)row + 1.0f - (float)IMG_S) / (float)IMG_S;
  const float xp = (2.0f * (float)(col0 + n) + 1.0f - (float)IMG_S) / (float)IMG_S;
  v2f a;
  a.x = hi ? 1.0f : xp;
  a.y = hi ? 0.0f : yp;

  // Best shifted-negated reciprocal depth (smaller = closer).
  // Init 9.99 = 10 - 1/FAR enforces zp < FAR; face -1 = background.
  float bq[8];
  int   bf[8];
#pragma unroll
  for (int r = 0; r < 8; ++r) { bq[r] = Q_INIT; bf[r] = -1; }

  const v8f czero = {};

#pragma unroll 2
  for (int tile = 0; tile < NTILES; ++tile) {
    const float* t = coef + tile * TSTR;

    v2f b0 = *(const v2f*)(t + 0 * 64 + 2 * lane);
    v2f b1 = *(const v2f*)(t + 1 * 64 + 2 * lane);
    v2f b2 = *(const v2f*)(t + 2 * 64 + 2 * lane);
    v2f b3 = *(const v2f*)(t + 3 * 64 + 2 * lane);

    v8f w0a = __builtin_amdgcn_wmma_f32_16x16x4_f32(false, a, false, b0,
                                                    (short)0, czero, false, false);
    v8f w1a = __builtin_amdgcn_wmma_f32_16x16x4_f32(false, a, false, b1,
                                                    (short)0, czero, false, false);
    v8f w2a = __builtin_amdgcn_wmma_f32_16x16x4_f32(false, a, false, b2,
                                                    (short)0, czero, false, false);
    v8f qqa = __builtin_amdgcn_wmma_f32_16x16x4_f32(false, a, false, b3,
                                                    (short)0, czero, false, false);

    const int faceIdx = (tile << 4) + n;

#pragma unroll
    for (int r = 0; r < 8; ++r) {
      float q = qqa[r];                                   // 10 - 1/zp
      float t3 = min3f(w0a[r], w1a[r], w2a[r]);           // inside
      float tv = min2f(t3, q);                            // + NEAR test
      bool upd = (tv > 0.0f) && (q < bq[r]);              // valid & closer
      if (upd) { bq[r] = q; bf[r] = faceIdx; }
    }
  }

  // Cross-lane min-q over the 16 face columns; xor masks < 16 keep the two
  // pixel half-groups independent. Ties -> lowest face index (unsigned
  // compare makes -1 = 0xffffffff lose).
#pragma unroll
  for (int r = 0; r < 8; ++r) {
    float q = bq[r];
    int   f = bf[r];
#pragma unroll
    for (int m = 8; m >= 1; m >>= 1) {
      float q2 = __shfl_xor(q, m, 32);
      int   f2 = __shfl_xor(f, m, 32);
      bool take = (q2 < q) || (q2 == q && (unsigned)f2 < (unsigned)f);
      if (take) { q = q2; f = f2; }
    }
    if (n == r) {
      out[row * IMG_S + col0 + (hi ? 8 : 0) + r] = f;
    }
  }
}

extern "C" void kernel_launch(void* const* d_in, const int* in_sizes, int n_in,
                              void* d_out, int out_size, void* d_ws, size_t ws_size,
                              hipStream_t stream) {
  (void)in_sizes; (void)n_in; (void)out_size; (void)ws_size;
  const float* faces = (const float*)d_in[0];   // (1, 4096, 3, 3) f32
  float* coef = (float*)d_ws;                   // 256 tiles * 256 floats = 256 KB
  int* out = (int*)d_out;                       // 256*256 int32 face indices

  coeff_kernel<<<NFACES / 256, 256, 0, stream>>>(faces, coef);
  raster_kernel<<<512, 256, 0, stream>>>(coef, out);   // 4096 waves
}